// RNN_46042049413728
// MI455X (gfx1250) — compile-verified
//
#include <hip/hip_runtime.h>
#include <math.h>

typedef float v2f __attribute__((ext_vector_type(2)));
typedef float v4f __attribute__((ext_vector_type(4)));
typedef float v8f __attribute__((ext_vector_type(8)));

#define HIDDEN 128
#define TSTEPS 28
#define BATCH  256
#define NSTEP  (BATCH * TSTEPS)   // 7168 sequential steps
#define DIN    28

// Hardware transcendental tanh (V_TANH_F32) when the builtin exists.
#if __has_builtin(__builtin_amdgcn_tanhf)
#define FAST_TANH(x) __builtin_amdgcn_tanhf(x)
#elif __has_builtin(__builtin_amdgcn_tanh_f32)
#define FAST_TANH(x) __builtin_amdgcn_tanh_f32(x)
#else
#define FAST_TANH(x) tanhf(x)
#endif

// ---------------------------------------------------------------------------
// Kernel 1: proj(7168x128) = x(7168x28) @ W_in^T(28x128), f32 WMMA 16x16x4.
// Block = 256 threads = 8 waves; blockIdx = 16-row tile, wave = 16-col tile.
// ---------------------------------------------------------------------------
__global__ __launch_bounds__(256) void esn_proj_wmma(const float* __restrict__ x,
                                                     const float* __restrict__ W_in,
                                                     float* __restrict__ proj) {
    const int lane = threadIdx.x & 31;
    const int wave = threadIdx.x >> 5;        // column tile 0..7
    const int row0 = blockIdx.x * 16;         // 448 row tiles
    const int col0 = wave * 16;
    const int m    = lane & 15;               // A row / B,D column within tile
    const int koff = (lane < 16) ? 0 : 2;     // K split across half-waves
    const int moff = (lane < 16) ? 0 : 8;     // M split for C/D

    v8f acc = {0.f, 0.f, 0.f, 0.f, 0.f, 0.f, 0.f, 0.f};
#pragma unroll
    for (int c = 0; c < 7; ++c) {             // K = 28 = 7 chunks of 4
        const int k = 4 * c + koff;
        v2f a = *(const v2f*)(x + (row0 + m) * DIN + k);
        v2f b = *(const v2f*)(W_in + (col0 + m) * DIN + k);
        acc = __builtin_amdgcn_wmma_f32_16x16x4_f32(false, a, false, b,
                                                    (short)0, acc, false, false);
    }
#pragma unroll
    for (int r = 0; r < 8; ++r)
        proj[(row0 + r + moff) * HIDDEN + col0 + m] = acc[r];
}

// ---------------------------------------------------------------------------
// Kernel 2: the sequential reservoir scan. One workgroup, 8 waves.
// Wave w owns h rows [16w,16w+16): 16x128 * 128x1 per step as 32 f32 WMMAs
// (4-way accumulator ILP). W_res resident in VGPRs; h ping-pongs in LDS
// (time loop unrolled x2 so buffer indices are compile-time constants);
// proj[t+1] is prefetched under the WMMA chain; activation is V_TANH_F32.
// ---------------------------------------------------------------------------
__global__ __launch_bounds__(256) void esn_scan_wmma(const float* __restrict__ proj,
                                                     const float* __restrict__ W_res,
                                                     float* __restrict__ h_last) {
    __shared__ float h_lds[2][HIDDEN];

    const int lane  = threadIdx.x & 31;
    const int wave  = threadIdx.x >> 5;       // row tile 0..7
    const int m     = lane & 15;
    const int koff  = (lane < 16) ? 0 : 2;
    const int moff  = (lane < 16) ? 0 : 8;
    const int rbase = wave * 16;

    // Preload A = W_res rows [rbase, rbase+16), 32 K-chunks, 64 VGPRs.
    v2f a[32];
#pragma unroll
    for (int c = 0; c < 32; ++c)
        a[c] = *(const v2f*)(W_res + (rbase + m) * HIDDEN + 4 * c + koff);

    if (threadIdx.x < HIDDEN) h_lds[0][threadIdx.x] = 0.f;   // h0 = 0
    __syncthreads();

    const v8f vzero = {0.f, 0.f, 0.f, 0.f, 0.f, 0.f, 0.f, 0.f};
    const float* pbase = proj + rbase + moff;                // per-lane proj base

    // Software-pipelined proj registers (current step's values).
    v4f c0 = *(const v4f*)(pbase);
    v4f c1 = *(const v4f*)(pbase + 4);

    for (int t = 0; t < NSTEP; t += 2) {
        // ================= even step: h_lds[0] -> h_lds[1] =================
        {
            v8f acc0;
#pragma unroll
            for (int r = 0; r < 4; ++r) { acc0[r] = c0[r]; acc0[r + 4] = c1[r]; }
            v8f acc1 = vzero, acc2 = vzero, acc3 = vzero;

            v2f b[32];
#pragma unroll
            for (int c = 0; c < 32; ++c)
                b[c] = *(const v2f*)&h_lds[0][4 * c + koff];

            // prefetch proj for step t+1 (overlaps the WMMA chain)
            v4f n0 = *(const v4f*)(pbase + (size_t)(t + 1) * HIDDEN);
            v4f n1 = *(const v4f*)(pbase + (size_t)(t + 1) * HIDDEN + 4);

#pragma unroll
            for (int c = 0; c < 32; c += 4) {
                acc0 = __builtin_amdgcn_wmma_f32_16x16x4_f32(false, a[c + 0], false, b[c + 0],
                                                             (short)0, acc0, false, false);
                acc1 = __builtin_amdgcn_wmma_f32_16x16x4_f32(false, a[c + 1], false, b[c + 1],
                                                             (short)0, acc1, false, false);
                acc2 = __builtin_amdgcn_wmma_f32_16x16x4_f32(false, a[c + 2], false, b[c + 2],
                                                             (short)0, acc2, false, false);
                acc3 = __builtin_amdgcn_wmma_f32_16x16x4_f32(false, a[c + 3], false, b[c + 3],
                                                             (short)0, acc3, false, false);
            }
            v8f s = (acc0 + acc1) + (acc2 + acc3);

            float hv[8];
#pragma unroll
            for (int r = 0; r < 8; ++r) hv[r] = FAST_TANH(s[r]);

            if (m == 0) {    // lanes 0 and 16 publish rows [0..7] / [8..15]
                v4f h0 = {hv[0], hv[1], hv[2], hv[3]};
                v4f h1 = {hv[4], hv[5], hv[6], hv[7]};
                *(v4f*)&h_lds[1][rbase + moff]     = h0;
                *(v4f*)&h_lds[1][rbase + moff + 4] = h1;
                // even t is never a batch end (batch ends at t%28==27, odd)
            }
            c0 = n0; c1 = n1;
        }
        __syncthreads();

        // ================= odd step: h_lds[1] -> h_lds[0] ==================
        {
            const int t1 = t + 1;
            v8f acc0;
#pragma unroll
            for (int r = 0; r < 4; ++r) { acc0[r] = c0[r]; acc0[r + 4] = c1[r]; }
            v8f acc1 = vzero, acc2 = vzero, acc3 = vzero;

            v2f b[32];
#pragma unroll
            for (int c = 0; c < 32; ++c)
                b[c] = *(const v2f*)&h_lds[1][4 * c + koff];

            // prefetch proj for step t+2 (clamped; last prefetch is unused)
            const int tn = (t + 2 < NSTEP) ? (t + 2) : (NSTEP - 1);
            v4f n0 = *(const v4f*)(pbase + (size_t)tn * HIDDEN);
            v4f n1 = *(const v4f*)(pbase + (size_t)tn * HIDDEN + 4);

#pragma unroll
            for (int c = 0; c < 32; c += 4) {
                acc0 = __builtin_amdgcn_wmma_f32_16x16x4_f32(false, a[c + 0], false, b[c + 0],
                                                             (short)0, acc0, false, false);
                acc1 = __builtin_amdgcn_wmma_f32_16x16x4_f32(false, a[c + 1], false, b[c + 1],
                                                             (short)0, acc1, false, false);
                acc2 = __builtin_amdgcn_wmma_f32_16x16x4_f32(false, a[c + 2], false, b[c + 2],
                                                             (short)0, acc2, false, false);
                acc3 = __builtin_amdgcn_wmma_f32_16x16x4_f32(false, a[c + 3], false, b[c + 3],
                                                             (short)0, acc3, false, false);
            }
            v8f s = (acc0 + acc1) + (acc2 + acc3);

            float hv[8];
#pragma unroll
            for (int r = 0; r < 8; ++r) hv[r] = FAST_TANH(s[r]);

            if (m == 0) {
                v4f h0 = {hv[0], hv[1], hv[2], hv[3]};
                v4f h1 = {hv[4], hv[5], hv[6], hv[7]};
                *(v4f*)&h_lds[0][rbase + moff]     = h0;
                *(v4f*)&h_lds[0][rbase + moff + 4] = h1;
                if ((t1 % TSTEPS) == (TSTEPS - 1)) {   // last step of batch item
                    const int bi = t1 / TSTEPS;
                    *(v4f*)(h_last + bi * HIDDEN + rbase + moff)     = h0;
                    *(v4f*)(h_last + bi * HIDDEN + rbase + moff + 4) = h1;
                }
            }
            c0 = n0; c1 = n1;
        }
        __syncthreads();
    }
}

// ---------------------------------------------------------------------------
// Kernel 3: out(256x10) = h_last(256x128) @ W_out^T + b_out. Tiny; VALU.
// ---------------------------------------------------------------------------
__global__ __launch_bounds__(256) void esn_readout(const float* __restrict__ h_last,
                                                   const float* __restrict__ W_out,
                                                   const float* __restrict__ b_out,
                                                   float* __restrict__ out) {
    const int t = blockIdx.x * blockDim.x + threadIdx.x;
    if (t >= BATCH * 10) return;
    const int b = t / 10;
    const int o = t % 10;
    float s = b_out[o];
#pragma unroll 4
    for (int k = 0; k < HIDDEN; ++k)
        s = fmaf(h_last[b * HIDDEN + k], W_out[o * HIDDEN + k], s);
    out[t] = s;
}

// ---------------------------------------------------------------------------
extern "C" void kernel_launch(void* const* d_in, const int* in_sizes, int n_in,
                              void* d_out, int out_size, void* d_ws, size_t ws_size,
                              hipStream_t stream) {
    const float* x     = (const float*)d_in[0];   // (256,28,28)
    const float* W_in  = (const float*)d_in[1];   // (128,28)
    const float* W_res = (const float*)d_in[2];   // (128,128)
    const float* W_out = (const float*)d_in[3];   // (10,128)
    const float* b_out = (const float*)d_in[4];   // (10,)
    float* out = (float*)d_out;                   // (256,10)

    float* proj   = (float*)d_ws;                  // 7168*128 f32 = 3.5 MB
    float* h_last = proj + (size_t)NSTEP * HIDDEN; // 256*128 f32

    esn_proj_wmma<<<NSTEP / 16, 256, 0, stream>>>(x, W_in, proj);
    esn_scan_wmma<<<1, 256, 0, stream>>>(proj, W_res, h_last);
    esn_readout<<<(BATCH * 10 + 255) / 256, 256, 0, stream>>>(h_last, W_out, b_out, out);
}